// King2019ModulationFilterbank_4148938408280
// MI455X (gfx1250) — compile-verified
//
#include <hip/hip_runtime.h>

typedef float v2f __attribute__((ext_vector_type(2)));
typedef float v8f __attribute__((ext_vector_type(8)));

#define NMOD   10
#define T_LEN  16000
#define NSEQ   1280            // 8*16*10 independent sequences
#define CHUNK  16
#define NCHUNK 1000            // T_LEN / CHUNK
#define GROUPS 63              // ceil(NCHUNK / 16) chunk-groups (superchunks) per sequence

// ---- workspace layout (in floats) ----
#define WS_A16   0                          // NMOD * 16   (A^16, 4x4 row-major)
#define WS_H     (WS_A16 + NMOD * 16)       // NMOD * 256  (16x16 impulse-response matrix)
#define WS_O     (WS_H + NMOD * 256)        // NMOD * 64   (16x4 initial-state response)
#define WS_A256  (WS_O + NMOD * 64)         // NMOD * 16   (A^256 for superchunk scan)
#define WS_STATE 4096                       // NSEQ * NCHUNK * 4 chunk states
#define WS_SUPER (WS_STATE + (size_t)NSEQ * NCHUNK * 4)   // NSEQ * GROUPS * 4
#define WS_FLOATS (WS_SUPER + (size_t)NSEQ * GROUPS * 4)

#define TILE_STRIDE 20          // padded chunk stride in LDS tile (gcd(20*? ,64) -> conflict-free)

struct Coef { float b00, b01, b02, a01, a02, b10, b11, b12, a11, a12; };

__device__ __forceinline__ Coef load_coef(const float* sos, int m) {
    const float* s0 = sos + m * 12;   // sos[m][0][:]
    const float* s1 = s0 + 6;         // sos[m][1][:]
    Coef c;
    c.b00 = s0[0]; c.b01 = s0[1]; c.b02 = s0[2]; c.a01 = s0[4]; c.a02 = s0[5];
    c.b10 = s1[0]; c.b11 = s1[1]; c.b12 = s1[2]; c.a11 = s1[4]; c.a12 = s1[5];
    return c;
}

// One cascade step on 4-dim state S = (s1a, s2a, s1b, s2b); returns output sample.
__device__ __forceinline__ float step4(const Coef& c, float S[4], float x) {
    float ya  = c.b00 * x  + S[0];
    float s1a = c.b01 * x  - c.a01 * ya + S[1];
    float s2a = c.b02 * x  - c.a02 * ya;
    float yb  = c.b10 * ya + S[2];
    float s1b = c.b11 * ya - c.a11 * yb + S[3];
    float s2b = c.b12 * ya - c.a12 * yb;
    S[0] = s1a; S[1] = s2a; S[2] = s1b; S[3] = s2b;
    return yb;
}

// S = A(4x4,row-major) * S + z
__device__ __forceinline__ void affine_step(const float A[16], float S[4],
                                            float z0, float z1, float z2, float z3) {
    float n0 = A[0]  * S[0] + A[1]  * S[1] + A[2]  * S[2] + A[3]  * S[3] + z0;
    float n1 = A[4]  * S[0] + A[5]  * S[1] + A[6]  * S[2] + A[7]  * S[3] + z1;
    float n2 = A[8]  * S[0] + A[9]  * S[1] + A[10] * S[2] + A[11] * S[3] + z2;
    float n3 = A[12] * S[0] + A[13] * S[1] + A[14] * S[2] + A[15] * S[3] + z3;
    S[0] = n0; S[1] = n1; S[2] = n2; S[3] = n3;
}

// ---------------------------------------------------------------------------
// Kernel 0: per-channel matrices: A^16, A^256, H (16x16 Toeplitz impulse
// response), O (16x4 zero-input state response). 10 tiny independent jobs.
// ---------------------------------------------------------------------------
__global__ void prep_kernel(const float* __restrict__ sos, float* __restrict__ ws) {
    int m = threadIdx.x;
    if (m >= NMOD) return;
    Coef c = load_coef(sos, m);

    // State-transition matrix A (columns = one step from each basis state, x=0)
    float A[16];
    for (int j = 0; j < 4; ++j) {
        float S[4] = {0.f, 0.f, 0.f, 0.f}; S[j] = 1.f;
        (void)step4(c, S, 0.f);
        for (int i = 0; i < 4; ++i) A[i * 4 + j] = S[i];
    }
    // Repeated squaring: A -> A^16 (4 squarings) -> A^256 (4 more)
    float Cur[16], Tmp[16];
    for (int i = 0; i < 16; ++i) Cur[i] = A[i];
    for (int sq = 0; sq < 8; ++sq) {
        for (int i = 0; i < 4; ++i)
            for (int j = 0; j < 4; ++j) {
                float acc = 0.f;
                for (int k = 0; k < 4; ++k) acc += Cur[i * 4 + k] * Cur[k * 4 + j];
                Tmp[i * 4 + j] = acc;
            }
        for (int i = 0; i < 16; ++i) Cur[i] = Tmp[i];
        if (sq == 3)
            for (int i = 0; i < 16; ++i) ws[WS_A16 + m * 16 + i] = Cur[i];
    }
    for (int i = 0; i < 16; ++i) ws[WS_A256 + m * 16 + i] = Cur[i];

    // Impulse response h[0..15] -> lower-triangular Toeplitz H
    float h[16];
    {
        float S[4] = {0.f, 0.f, 0.f, 0.f};
        for (int t = 0; t < 16; ++t) h[t] = step4(c, S, (t == 0) ? 1.f : 0.f);
    }
    for (int i = 0; i < 16; ++i)
        for (int j = 0; j < 16; ++j)
            ws[WS_H + m * 256 + i * 16 + j] = (i >= j) ? h[i - j] : 0.f;

    // O[t][j] = output at step t from initial state e_j, zero input
    for (int j = 0; j < 4; ++j) {
        float S[4] = {0.f, 0.f, 0.f, 0.f}; S[j] = 1.f;
        for (int t = 0; t < 16; ++t)
            ws[WS_O + m * 64 + t * 4 + j] = step4(c, S, 0.f);
    }
}

// ---------------------------------------------------------------------------
// Kernel 1: zero-state final state of every 16-sample chunk. 1.28M threads.
// ---------------------------------------------------------------------------
__global__ __launch_bounds__(256)
void chunk_state_kernel(const float* __restrict__ x, const float* __restrict__ sos,
                        float* __restrict__ ws) {
    int tid = blockIdx.x * blockDim.x + threadIdx.x;
    if (tid >= NSEQ * NCHUNK) return;
    int p   = tid % NCHUNK;
    int seq = tid / NCHUNK;
    int m   = seq % NMOD;
    Coef c = load_coef(sos, m);
    const float* xp = x + (size_t)(seq / NMOD) * T_LEN + p * CHUNK;
    float4 xv[4];
    #pragma unroll
    for (int q = 0; q < 4; ++q) xv[q] = ((const float4*)xp)[q];
    const float* xs = (const float*)xv;
    float S[4] = {0.f, 0.f, 0.f, 0.f};
    #pragma unroll
    for (int t = 0; t < CHUNK; ++t) (void)step4(c, S, xs[t]);
    float* st = ws + WS_STATE + ((size_t)seq * NCHUNK + p) * 4;
    st[0] = S[0]; st[1] = S[1]; st[2] = S[2]; st[3] = S[3];
}

// ---------------------------------------------------------------------------
// Kernel 2a: superchunk reduce — zero-state result of 16 consecutive chunk
// affine maps. 80,640 threads, serial depth 16.
// ---------------------------------------------------------------------------
__global__ __launch_bounds__(256)
void super_reduce_kernel(float* __restrict__ ws) {
    int tid = blockIdx.x * blockDim.x + threadIdx.x;
    if (tid >= NSEQ * GROUPS) return;
    int s   = tid % GROUPS;
    int seq = tid / GROUPS;
    int m   = seq % NMOD;
    float A[16];
    #pragma unroll
    for (int i = 0; i < 16; ++i) A[i] = ws[WS_A16 + m * 16 + i];
    const float* st = ws + WS_STATE + (size_t)seq * NCHUNK * 4;
    float S[4] = {0.f, 0.f, 0.f, 0.f};
    #pragma unroll
    for (int j = 0; j < 16; ++j) {
        int p = s * 16 + j;
        if (p < NCHUNK)
            affine_step(A, S, st[p*4+0], st[p*4+1], st[p*4+2], st[p*4+3]);
    }
    float* sp = ws + WS_SUPER + (size_t)tid * 4;
    sp[0] = S[0]; sp[1] = S[1]; sp[2] = S[2]; sp[3] = S[3];
}

// ---------------------------------------------------------------------------
// Kernel 2b: exclusive scan over 63 superchunks per sequence with A^256.
// 1280 threads, serial depth 63 (was 1000).
// ---------------------------------------------------------------------------
__global__ void super_scan_kernel(float* __restrict__ ws) {
    int seq = blockIdx.x * blockDim.x + threadIdx.x;
    if (seq >= NSEQ) return;
    int m = seq % NMOD;
    float A[16];
    #pragma unroll
    for (int i = 0; i < 16; ++i) A[i] = ws[WS_A256 + m * 16 + i];
    float* sp = ws + WS_SUPER + (size_t)seq * GROUPS * 4;
    float S[4] = {0.f, 0.f, 0.f, 0.f};
    for (int s = 0; s < GROUPS; ++s) {
        float z0 = sp[s*4+0], z1 = sp[s*4+1], z2 = sp[s*4+2], z3 = sp[s*4+3];
        sp[s*4+0] = S[0]; sp[s*4+1] = S[1]; sp[s*4+2] = S[2]; sp[s*4+3] = S[3];
        affine_step(A, S, z0, z1, z2, z3);
    }
}

// ---------------------------------------------------------------------------
// Kernel 2c: fill in per-chunk entry states within each superchunk.
// 80,640 threads, serial depth 16. In place: chunk slot p ends up holding the
// state *entering* chunk p.
// ---------------------------------------------------------------------------
__global__ __launch_bounds__(256)
void chunk_fill_kernel(float* __restrict__ ws) {
    int tid = blockIdx.x * blockDim.x + threadIdx.x;
    if (tid >= NSEQ * GROUPS) return;
    int s   = tid % GROUPS;
    int seq = tid / GROUPS;
    int m   = seq % NMOD;
    float A[16];
    #pragma unroll
    for (int i = 0; i < 16; ++i) A[i] = ws[WS_A16 + m * 16 + i];
    const float* sp = ws + WS_SUPER + (size_t)tid * 4;
    float S[4] = { sp[0], sp[1], sp[2], sp[3] };
    float* st = ws + WS_STATE + (size_t)seq * NCHUNK * 4;
    #pragma unroll
    for (int j = 0; j < 16; ++j) {
        int p = s * 16 + j;
        if (p < NCHUNK) {
            float z0 = st[p*4+0], z1 = st[p*4+1], z2 = st[p*4+2], z3 = st[p*4+3];
            st[p*4+0] = S[0]; st[p*4+1] = S[1]; st[p*4+2] = S[2]; st[p*4+3] = S[3];
            affine_step(A, S, z0, z1, z2, z3);
        }
    }
}

// ---------------------------------------------------------------------------
// Kernel 3: per wave, 16 chunks = 16 columns of a 16x16 tile:
//   Y = H(16x16) * X(16x16) + O(16x4) * S_in(4x16)
// as 5 accumulating V_WMMA_F32_16X16X4_F32 ops. All global traffic is
// coalesced b128; the lane-strided transposes go through LDS:
//   - H/O for all 10 channels staged per block (12.8 KB)
//   - per-wave padded x/y tile (stride 20 floats -> conflict-free banks)
// Fragment layouts per ISA 7.12.2: A/B K = vgpr + 2*(lane>=16), M/N = lane&15;
// D: M = vgpr + 8*(lane>=16), N = lane&15.
// ---------------------------------------------------------------------------
__global__ __launch_bounds__(256)
void wmma_out_kernel(const float* __restrict__ x, const float* __restrict__ ws,
                     float* __restrict__ out) {
    __shared__ float sHO[NMOD * 320];                 // H (NMOD*256) then O (NMOD*64)
    __shared__ float tile[8][16 * TILE_STRIDE];       // per-wave padded x/y tile

    // Block-cooperative, coalesced stage of all channel matrices.
    for (int i = threadIdx.x; i < NMOD * 320; i += 256)
        sHO[i] = ws[WS_H + i];
    __syncthreads();

    int wave = (blockIdx.x * blockDim.x + threadIdx.x) >> 5;
    int lane = threadIdx.x & 31;
    if (wave >= NSEQ * GROUPS) wave = NSEQ * GROUPS - 1;   // exact grid; clamp for safety

    int seq  = wave / GROUPS;
    int g    = wave % GROUPS;
    int m    = seq % NMOD;
    int half = lane >> 4;
    int col  = lane & 15;
    int chunk  = g * 16 + col;
    int cclamp = (chunk < NCHUNK) ? chunk : (NCHUNK - 1);
    int kbase  = 2 * half;
    int tbase  = g * 256;

    const float* xp = x + (size_t)(seq / NMOD) * T_LEN;
    const float* sHm = sHO + m * 256;
    const float* sOm = sHO + NMOD * 256 + m * 64;
    const float* st  = ws + WS_STATE + (size_t)seq * NCHUNK * 4;
    float* tw = tile[threadIdx.x >> 5];

    // ---- stage this wave's 256 input samples: coalesced global -> padded LDS ----
    int ctile = lane >> 1;               // chunk index this lane stages
    int koff  = (lane & 1) * 8;          // first/second half of that chunk
    if (g < GROUPS - 1) {                // full group: 2 x b128 per lane
        const float* xg = xp + tbase + lane * 8;
        float4 u0 = ((const float4*)xg)[0];
        float4 u1 = ((const float4*)xg)[1];
        *(float4*)(tw + ctile * TILE_STRIDE + koff)     = u0;
        *(float4*)(tw + ctile * TILE_STRIDE + koff + 4) = u1;
    } else {                             // last (partial) group: clamped scalar loads
        #pragma unroll
        for (int j = 0; j < 8; ++j) {
            int t = tbase + lane * 8 + j;
            tw[ctile * TILE_STRIDE + koff + j] = xp[(t < T_LEN) ? t : (T_LEN - 1)];
        }
    }

    // ---- fragments ----
    v2f a0, a1, a2, a3, of, b0, b1, b2, b3, sf;
    int row = col;  // A-operand: lane&15 is the output-row M
    a0.x = sHm[row * 16 +  0 + kbase]; a0.y = sHm[row * 16 +  1 + kbase];
    a1.x = sHm[row * 16 +  4 + kbase]; a1.y = sHm[row * 16 +  5 + kbase];
    a2.x = sHm[row * 16 +  8 + kbase]; a2.y = sHm[row * 16 +  9 + kbase];
    a3.x = sHm[row * 16 + 12 + kbase]; a3.y = sHm[row * 16 + 13 + kbase];
    of.x = sOm[row * 4 + kbase];       of.y = sOm[row * 4 + 1 + kbase];

    const float* tc = tw + col * TILE_STRIDE;   // B-operand: lane&15 is the column N
    b0.x = tc[ 0 + kbase]; b0.y = tc[ 1 + kbase];
    b1.x = tc[ 4 + kbase]; b1.y = tc[ 5 + kbase];
    b2.x = tc[ 8 + kbase]; b2.y = tc[ 9 + kbase];
    b3.x = tc[12 + kbase]; b3.y = tc[13 + kbase];

    const float* sc = st + (size_t)cclamp * 4;
    sf.x = sc[kbase]; sf.y = sc[kbase + 1];

    v8f acc = {};
    // 8 args: (neg_a, A, neg_b, B, c_mod, C, reuse_a, reuse_b)
    acc = __builtin_amdgcn_wmma_f32_16x16x4_f32(false, of, false, sf, (short)0, acc, false, false);
    acc = __builtin_amdgcn_wmma_f32_16x16x4_f32(false, a0, false, b0, (short)0, acc, false, false);
    acc = __builtin_amdgcn_wmma_f32_16x16x4_f32(false, a1, false, b1, (short)0, acc, false, false);
    acc = __builtin_amdgcn_wmma_f32_16x16x4_f32(false, a2, false, b2, (short)0, acc, false, false);
    acc = __builtin_amdgcn_wmma_f32_16x16x4_f32(false, a3, false, b3, (short)0, acc, false, false);

    if (g + 1 < GROUPS)
        __builtin_prefetch(xp + (size_t)(g + 1) * 256, 0, 1);  // global_prefetch_b8

    // ---- D tile -> LDS (transpose to time-major), then coalesced store ----
    float* ty = tw + col * TILE_STRIDE + 8 * half;  // rows r..r+7 of this column
    float4 lo, hi;
    lo.x = acc[0]; lo.y = acc[1]; lo.z = acc[2]; lo.w = acc[3];
    hi.x = acc[4]; hi.y = acc[5]; hi.z = acc[6]; hi.w = acc[7];
    *(float4*)(ty)     = lo;
    *(float4*)(ty + 4) = hi;
    __syncthreads();   // order LDS writes before cross-lane readback

    float4 r0 = *(const float4*)(tw + ctile * TILE_STRIDE + koff);
    float4 r1 = *(const float4*)(tw + ctile * TILE_STRIDE + koff + 4);
    int tglob = tbase + lane * 8;
    if (tglob < T_LEN) {
        float* op = out + (size_t)seq * T_LEN + tglob;
        ((float4*)op)[0] = r0;
        ((float4*)op)[1] = r1;
    }
}

// ---------------------------------------------------------------------------
// Fallback: one thread per sequence, plain recurrence (only if d_ws too small).
// ---------------------------------------------------------------------------
__global__ void naive_kernel(const float* __restrict__ x, const float* __restrict__ sos,
                             float* __restrict__ out) {
    int seq = blockIdx.x * blockDim.x + threadIdx.x;
    if (seq >= NSEQ) return;
    Coef c = load_coef(sos, seq % NMOD);
    const float* xp = x + (size_t)(seq / NMOD) * T_LEN;
    float* yp = out + (size_t)seq * T_LEN;
    float S[4] = {0.f, 0.f, 0.f, 0.f};
    for (int t = 0; t < T_LEN; ++t) yp[t] = step4(c, S, xp[t]);
}

extern "C" void kernel_launch(void* const* d_in, const int* in_sizes, int n_in,
                              void* d_out, int out_size, void* d_ws, size_t ws_size,
                              hipStream_t stream) {
    const float* x   = (const float*)d_in[0];   // (8,16,16000) f32
    const float* sos = (const float*)d_in[1];   // (10,2,6) f32
    float* out = (float*)d_out;                 // (8,16,10,16000) f32
    float* ws  = (float*)d_ws;

    if (ws_size < WS_FLOATS * sizeof(float)) {
        naive_kernel<<<(NSEQ + 127) / 128, 128, 0, stream>>>(x, sos, out);
        return;
    }

    prep_kernel<<<1, 32, 0, stream>>>(sos, ws);

    int total_chunks = NSEQ * NCHUNK;                       // 1,280,000
    chunk_state_kernel<<<(total_chunks + 255) / 256, 256, 0, stream>>>(x, sos, ws);

    int total_supers = NSEQ * GROUPS;                       // 80,640
    super_reduce_kernel<<<(total_supers + 255) / 256, 256, 0, stream>>>(ws);
    super_scan_kernel<<<(NSEQ + 127) / 128, 128, 0, stream>>>(ws);
    chunk_fill_kernel<<<(total_supers + 255) / 256, 256, 0, stream>>>(ws);

    int threads = total_supers * 32;                        // 2,580,480 (exact waves)
    wmma_out_kernel<<<(threads + 255) / 256, 256, 0, stream>>>(x, ws, out);
}